// PRGNN_27376121545190
// MI455X (gfx1250) — compile-verified
//
#include <hip/hip_runtime.h>
#include <hip/hip_bf16.h>

typedef __attribute__((ext_vector_type(2))) float v2f;
typedef __attribute__((ext_vector_type(8))) float v8f;

#define NNODES 50000          // multiple of 16: GEMM tiles cover rows exactly
#define NEDGES 200000
#define NSEG   512
#define NPAIR  1024
#define UCOLS  320            // 256 (8 edge kernels) + 32 (bk term) + 32 (root)
#define LDSP   33             // padded row stride in LDS to dodge bank conflicts

// ---------------------------------------------------------------------------
// U = A(n x 32, row-stride lda) @ Wbig(32 x 320), Wbig assembled in LDS from
// Wk (8x1024), bk (1024), root (32x32).  f32 WMMA 16x16x4, K looped 8 times.
// REQUIRES: nrows % 16 == 0 (stores are unguarded within an active wave tile).
// ---------------------------------------------------------------------------
__global__ __launch_bounds__(256) void ecc_node_gemm(
    const float* __restrict__ Ain, int lda,
    const float* __restrict__ Wk, const float* __restrict__ bk,
    const float* __restrict__ root,
    float* __restrict__ U, int nrows)
{
    __shared__ float Wlds[UCOLS * LDSP];   // Wlds[c*LDSP + i] = Wbig[i][c]
    const int t = threadIdx.x;

    for (int idx = t; idx < UCOLS * 32; idx += 256) {
        const int c = idx >> 5;
        const int i = idx & 31;
        float w;
        if (c < 256) {
            const int s = c >> 5, o = c & 31;
            w = Wk[s * 1024 + i * 32 + o];
        } else if (c < 288) {
            w = bk[i * 32 + (c - 256)];
        } else {
            w = root[i * 32 + (c - 288)];
        }
        Wlds[c * LDSP + i] = w;
    }
    __syncthreads();

    const int wave    = t >> 5;
    const int lane    = t & 31;
    const int halfsel = lane >> 4;   // 0: lanes 0-15 (K=0,1) | 1: lanes 16-31 (K=2,3)
    const int lrow    = lane & 15;
    const long m0     = (long)(blockIdx.x * 8 + wave) * 16;
    if (m0 >= nrows) return;         // whole-wave exit: EXEC stays all-ones for WMMA

    // A fragments for all eight K-steps: lane holds A[lrow][4k + 2*halfsel + {0,1}]
    v2f a[8];
    const float* ap = Ain + (m0 + lrow) * (long)lda + halfsel * 2;
#pragma unroll
    for (int k = 0; k < 8; ++k) {
        a[k].x = ap[4 * k + 0];
        a[k].y = ap[4 * k + 1];
    }

    // store base for this lane: row m0 + 8*halfsel (+r), col c*16 + lrow
    float* Ubase = U + (m0 + (long)halfsel * 8) * UCOLS + lrow;

    for (int c = 0; c < UCOLS / 16; ++c) {
        const float* bp = &Wlds[(c * 16 + lrow) * LDSP + halfsel * 2];
        v8f acc = {0.f, 0.f, 0.f, 0.f, 0.f, 0.f, 0.f, 0.f};
#pragma unroll
        for (int k = 0; k < 8; ++k) {
            v2f b;
            b.x = bp[4 * k + 0];
            b.y = bp[4 * k + 1];
            acc = __builtin_amdgcn_wmma_f32_16x16x4_f32(
                false, a[k], false, b, (short)0, acc, false, false);
        }
        float* up = Ubase + c * 16;
#pragma unroll
        for (int r = 0; r < 8; ++r)
            up[(long)r * UCOLS] = acc[r];
    }
}

// ---------------------------------------------------------------------------
// Per-edge: msg_o = U[src,256+o] + sum_s e[edge,s]*U[src,s*32+o]; scatter-add.
// One edge per wave, lane == output channel.
// ---------------------------------------------------------------------------
__global__ __launch_bounds__(256) void ecc_edge(
    const float* __restrict__ U, const float* __restrict__ ew,
    const int* __restrict__ src, const int* __restrict__ tgt,
    float* __restrict__ agg, int nE)
{
    const int gid  = blockIdx.x * 256 + threadIdx.x;
    const int edge = gid >> 5;
    const int o    = gid & 31;
    if (edge >= nE) return;

    const int  sn   = src[edge];
    const int  tn   = tgt[edge];
    const float* Ur = U + (long)sn * UCOLS;
    __builtin_prefetch(Ur + o, 0, 0);

    float msg = Ur[256 + o];                       // bk contribution
    const float* ep = ew + (long)edge * 8;
#pragma unroll
    for (int s = 0; s < 8; ++s)
        msg = fmaf(ep[s], Ur[s * 32 + o], msg);

    atomicAdd(&agg[(long)tn * 32 + o], msg);
}

// ---------------------------------------------------------------------------
// h = relu(agg + U[:,288:320] + b); optional trunc (layer 2).
// ---------------------------------------------------------------------------
__global__ __launch_bounds__(256) void ecc_finalize(
    const float* __restrict__ agg, const float* __restrict__ U,
    const float* __restrict__ bias, float* __restrict__ h,
    int nrows, int do_trunc)
{
    const long gid = (long)blockIdx.x * 256 + threadIdx.x;
    if (gid >= (long)nrows * 32) return;
    const long node = gid >> 5;
    const int  o    = (int)(gid & 31);
    float v = agg[gid] + U[node * UCOLS + 288 + o] + bias[o];
    v = fmaxf(v, 0.f);
    if (do_trunc) v = truncf(v);
    h[gid] = v;
}

__global__ __launch_bounds__(256) void zero_f32(float* __restrict__ p, long n)
{
    const long gid = (long)blockIdx.x * 256 + threadIdx.x;
    if (gid < n) p[gid] = 0.f;
}

__global__ __launch_bounds__(256) void seg_pool(
    const float* __restrict__ h, const int* __restrict__ seg,
    float* __restrict__ pooled, int nrows)
{
    const long gid = (long)blockIdx.x * 256 + threadIdx.x;
    if (gid >= (long)nrows * 32) return;
    const long node = gid >> 5;
    const int  o    = (int)(gid & 31);
    atomicAdd(&pooled[(long)seg[node] * 32 + o], h[gid]);
}

__global__ __launch_bounds__(256) void dense_head(
    const float* __restrict__ pooled, const float* __restrict__ Wd,
    const float* __restrict__ bd, float* __restrict__ outg, int ng)
{
    const int g = blockIdx.x * 256 + threadIdx.x;
    if (g >= ng) return;
    float s = bd[0];
#pragma unroll
    for (int i = 0; i < 32; ++i)
        s = fmaf(pooled[(long)g * 32 + i], Wd[i], s);
    outg[g] = fmaxf(s, 0.f);
}

__global__ __launch_bounds__(256) void pair_diff(
    const float* __restrict__ outg, const int* __restrict__ ia,
    const int* __restrict__ ib, float* __restrict__ out, int np)
{
    const int p = blockIdx.x * 256 + threadIdx.x;
    if (p >= np) return;
    out[p] = outg[ib[p]] - outg[ia[p]];
}

// ---------------------------------------------------------------------------
extern "C" void kernel_launch(void* const* d_in, const int* in_sizes, int n_in,
                              void* d_out, int out_size, void* d_ws, size_t ws_size,
                              hipStream_t stream)
{
    const float* x     = (const float*)d_in[0];   // N x 33
    const float* e     = (const float*)d_in[1];   // E x 8
    const int*   srcs  = (const int*)  d_in[2];
    const int*   tgts  = (const int*)  d_in[3];
    const int*   seg   = (const int*)  d_in[4];
    const int*   idx_a = (const int*)  d_in[5];
    const int*   idx_b = (const int*)  d_in[6];
    const float* Wk1   = (const float*)d_in[7];
    const float* bk1   = (const float*)d_in[8];
    const float* root1 = (const float*)d_in[9];
    const float* b1    = (const float*)d_in[10];
    const float* Wk2   = (const float*)d_in[11];
    const float* bk2   = (const float*)d_in[12];
    const float* root2 = (const float*)d_in[13];
    const float* b2    = (const float*)d_in[14];
    const float* Wd    = (const float*)d_in[15];
    const float* bd    = (const float*)d_in[16];
    float* out = (float*)d_out;

    // workspace carve-up
    char* ws = (char*)d_ws;
    float* U      = (float*)ws;                       ws += (size_t)NNODES * UCOLS * 4; // 64 MB
    float* agg    = (float*)ws;                       ws += (size_t)NNODES * 32 * 4;    // 6.4 MB
    float* h      = (float*)ws;                       ws += (size_t)NNODES * 32 * 4;    // 6.4 MB
    float* pooled = (float*)ws;                       ws += (size_t)NSEG * 32 * 4;
    float* outg   = (float*)ws;

    const int gemm_blocks = (NNODES + 16 * 8 - 1) / (16 * 8);     // 391
    const int edge_blocks = (NEDGES * 32 + 255) / 256;            // 25000
    const int node_blocks = (NNODES * 32 + 255) / 256;            // 6250

    // ---- layer 1 (reads x with row stride 33, dropping last column) ----
    zero_f32<<<node_blocks, 256, 0, stream>>>(agg, (long)NNODES * 32);
    ecc_node_gemm<<<gemm_blocks, 256, 0, stream>>>(x, 33, Wk1, bk1, root1, U, NNODES);
    ecc_edge<<<edge_blocks, 256, 0, stream>>>(U, e, srcs, tgts, agg, NEDGES);
    ecc_finalize<<<node_blocks, 256, 0, stream>>>(agg, U, b1, h, NNODES, 0);

    // ---- layer 2 (+ trunc fused into finalize) ----
    zero_f32<<<node_blocks, 256, 0, stream>>>(agg, (long)NNODES * 32);
    ecc_node_gemm<<<gemm_blocks, 256, 0, stream>>>(h, 32, Wk2, bk2, root2, U, NNODES);
    ecc_edge<<<edge_blocks, 256, 0, stream>>>(U, e, srcs, tgts, agg, NEDGES);
    ecc_finalize<<<node_blocks, 256, 0, stream>>>(agg, U, b2, h, NNODES, 1);

    // ---- pooling + head + pairs ----
    zero_f32<<<(NSEG * 32 + 255) / 256, 256, 0, stream>>>(pooled, (long)NSEG * 32);
    seg_pool<<<node_blocks, 256, 0, stream>>>(h, seg, pooled, NNODES);
    dense_head<<<(NSEG + 255) / 256, 256, 0, stream>>>(pooled, Wd, bd, outg, NSEG);
    pair_diff<<<(NPAIR + 255) / 256, 256, 0, stream>>>(outg, idx_a, idx_b, out, NPAIR);
}